// GridLinearModel_83992380440947
// MI455X (gfx1250) — compile-verified
//
#include <hip/hip_runtime.h>
#include <stdint.h>

// ---------------------------------------------------------------------------
// GridLinearModel fused loss for MI455X (gfx1250, wave32)
//   proj kernel : D = relu(X @ W^T + b) * mask, then collapse to 3 classes
//                 via W1 -> proj[b, i, 3].  Uses V_WMMA_F32_16X16X4_F32 and
//                 TDM tensor_load_to_lds double-buffered tiles.
//   loss kernel : log_softmax over 3 logits per grid cell + NLL mean.
// Workspace layout (floats): d_proj[8*1024*3] | e_proj[8*1024*3] | partials[8192*2]
//   = 256 KB total.
// ---------------------------------------------------------------------------

typedef __attribute__((ext_vector_type(2))) float v2f;
typedef __attribute__((ext_vector_type(8))) float v8f;
typedef __attribute__((ext_vector_type(4))) unsigned int v4u;
typedef __attribute__((ext_vector_type(8))) int v8i;
typedef __attribute__((ext_vector_type(4))) int v4i;

#define F_DIM 768
#define H_DIM 1024
#define PADS 771 // 768 dwords + 1 pad dword per 256 (TDM pad) -> stride%64 = 3 banks

#if defined(__has_builtin)
#if __has_builtin(__builtin_amdgcn_tensor_load_to_lds)
#define HAVE_TDM 1
#endif
#endif
#ifndef HAVE_TDM
#define HAVE_TDM 0
#endif

#if HAVE_TDM
// Build a D# per CDNA5 ISA ch.8 and issue TENSOR_LOAD_TO_LDS for a 2D f32
// tile (rows x width, contiguous rows in memory) with LDS padding of 1 dword
// every 256 dwords (pad_interval code 7, pad_amount code 0).
// This toolchain's builtin is the 6-operand form:
//   (uint32x4 g0, int32x8 g1, int32x4 g2, int32x4 g3, int32x8 extra, int cpol)
__device__ __forceinline__ void tdm_load_tile_f32(uint32_t lds_addr,
                                                  const float* gptr,
                                                  uint32_t width, uint32_t rows) {
  uint64_t ga = (uint64_t)(uintptr_t)gptr;
  v4u g0;
  g0.x = 1u;                                                  // count=1, user desc
  g0.y = lds_addr;                                            // LDS byte address
  g0.z = (uint32_t)ga;                                        // global addr lo
  g0.w = ((uint32_t)(ga >> 32) & 0x01FFFFFFu) | (2u << 30);   // addr[56:32], type=2
  v8i g1;
  g1[0] = (int)((2u << 16) | (1u << 20) | (7u << 22));        // 4B elems, pad_en, 256dw, +1dw
  g1[1] = (int)((width & 0xFFFFu) << 16);                     // tensor_dim0[15:0]
  g1[2] = (int)(((width >> 16) & 0xFFFFu) | ((rows & 0xFFFFu) << 16)); // dim0 hi | dim1 lo
  g1[3] = (int)(((rows >> 16) & 0xFFFFu) | ((width & 0xFFFFu) << 16)); // dim1 hi | tile_dim0
  g1[4] = (int)(rows & 0xFFFFu);                              // tile_dim1 (tile_dim2=0)
  g1[5] = (int)width;                                         // tensor_dim0_stride lo
  g1[6] = 0;
  g1[7] = 0;
  v4i g2 = {0, 0, 0, 0};   // 2D tile: tensor_dim2.. unused
  v4i g3 = {0, 0, 0, 0};
  v8i g4 = {0, 0, 0, 0, 0, 0, 0, 0};
  __builtin_amdgcn_tensor_load_to_lds(g0, g1, g2, g3, g4, 0);
}
#else
// Fallback: cooperative copy with the same padded LDS layout.
__device__ __forceinline__ void coop_copy_tile(float* dst, const float* src,
                                               int rows, int tid, int nthr) {
  for (int r = 0; r < rows; ++r)
    for (int k = tid; k < F_DIM; k += nthr)
      dst[r * PADS + k + (k >> 8)] = src[r * F_DIM + k];
}
#endif

__device__ __forceinline__ v8f wmma_f32_k4(v2f a, v2f b, v8f c) {
  return __builtin_amdgcn_wmma_f32_16x16x4_f32(false, a, false, b, (short)0, c,
                                               false, false);
}

// grid: 256 blocks (8 batches * 32 row-tiles of 32), block: 128 threads (4 waves)
// wave w: rblk = w&1 (16-row block), cblk = w>>1 (16-col block of the 32-h slab)
__global__ __launch_bounds__(128) void proj_fused_kernel(
    const float* __restrict__ X,    // [8,1024,768]
    const float* __restrict__ W,    // [1024,768]
    const float* __restrict__ bias, // [1024]
    const float* __restrict__ W1,   // [3,2048]
    int w1off,                      // 0 (decoder half) or 1024 (encoder half)
    const float* __restrict__ mask, // [8,1024]
    float* __restrict__ proj)       // [8,1024,3]
{
  __shared__ float smA[32 * PADS];
  __shared__ float smW[2][32 * PADS];
  __shared__ float stage[2][32][3];

  const int b    = blockIdx.x >> 5;
  const int row0 = (blockIdx.x & 31) * 32;
  const int tid  = threadIdx.x;
  const int wave = tid >> 5;
  const int lane = tid & 31;
  const int half = lane >> 4;
  const int n15  = lane & 15;
  const int rblk = wave & 1;
  const int cblk = wave >> 1;

#if HAVE_TDM
  if (wave == 0) {
    tdm_load_tile_f32((uint32_t)(uintptr_t)smA,
                      X + ((size_t)b * H_DIM + row0) * F_DIM, F_DIM, 32);
    tdm_load_tile_f32((uint32_t)(uintptr_t)&smW[0][0], W, F_DIM, 32);
  }
#else
  coop_copy_tile(smA, X + ((size_t)b * H_DIM + row0) * F_DIM, 32, tid, 128);
#endif

  const float* Arow = smA + (rblk * 16 + n15) * PADS;

  float p[8][3];
#pragma unroll
  for (int r = 0; r < 8; ++r) {
    p[r][0] = 0.f; p[r][1] = 0.f; p[r][2] = 0.f;
  }

  for (int it = 0; it < 32; ++it) {
#if HAVE_TDM
    if (wave == 0) __builtin_amdgcn_s_wait_tensorcnt(0);
    __syncthreads(); // buffer it&1 valid; prior compute on (it+1)&1 done
    if (wave == 0 && it < 31)
      tdm_load_tile_f32((uint32_t)(uintptr_t)&smW[(it + 1) & 1][0],
                        W + (size_t)(it + 1) * 32 * F_DIM, F_DIM, 32);
    const float* Wrow = &smW[it & 1][0] + (cblk * 16 + n15) * PADS;
#else
    __syncthreads();
    coop_copy_tile(&smW[0][0], W + (size_t)it * 32 * F_DIM, 32, tid, 128);
    __syncthreads();
    const float* Wrow = &smW[0][0] + (cblk * 16 + n15) * PADS;
#endif

    v8f acc0 = {};
    v8f acc1 = {};
#pragma unroll 4
    for (int k0 = 0; k0 < F_DIM; k0 += 8) {
      const int x0 = k0 + 2 * half;
      const int o0 = x0 + (x0 >> 8);
      v2f a0 = {Arow[o0], Arow[o0 + 1]};
      v2f b0 = {Wrow[o0], Wrow[o0 + 1]};
      acc0 = wmma_f32_k4(a0, b0, acc0);
      const int x1 = x0 + 4;
      const int o1 = x1 + (x1 >> 8);
      v2f a1 = {Arow[o1], Arow[o1 + 1]};
      v2f b1v = {Wrow[o1], Wrow[o1 + 1]};
      acc1 = wmma_f32_k4(a1, b1v, acc1);
    }

    const int h = it * 32 + cblk * 16 + n15;
    const float bh  = bias[h];
    const float w10 = W1[w1off + h];
    const float w11 = W1[2048 + w1off + h];
    const float w12 = W1[4096 + w1off + h];
#pragma unroll
    for (int r = 0; r < 8; ++r) {
      float v = acc0[r] + acc1[r] + bh;
      v = v > 0.f ? v : 0.f; // relu
      p[r][0] += v * w10;
      p[r][1] += v * w11;
      p[r][2] += v * w12;
    }
  }

  // Reduce the 16 N-lanes within each half-wave (rows live per C-layout:
  // lanes 0-15 -> M=r, lanes 16-31 -> M=8+r).
#pragma unroll
  for (int m = 1; m <= 8; m <<= 1) {
#pragma unroll
    for (int r = 0; r < 8; ++r) {
      p[r][0] += __shfl_xor(p[r][0], m, 32);
      p[r][1] += __shfl_xor(p[r][1], m, 32);
      p[r][2] += __shfl_xor(p[r][2], m, 32);
    }
  }
  if (n15 == 0) {
#pragma unroll
    for (int r = 0; r < 8; ++r) {
      const int rowl = rblk * 16 + half * 8 + r;
      stage[cblk][rowl][0] = p[r][0];
      stage[cblk][rowl][1] = p[r][1];
      stage[cblk][rowl][2] = p[r][2];
    }
  }
  __syncthreads();
  if (tid < 96) {
    const int rowl = tid / 3, c = tid % 3;
    const int row = row0 + rowl;
    const float v =
        (stage[0][rowl][c] + stage[1][rowl][c]) * mask[b * H_DIM + row];
    proj[((size_t)b * H_DIM + row) * 3 + c] = v;
  }
}

// grid: 8192 blocks = (b,d) pairs; 256 threads iterate over e.
__global__ __launch_bounds__(256) void grid_nll_kernel(
    const float* __restrict__ dproj, const float* __restrict__ eproj,
    const float* __restrict__ b1, const int* __restrict__ labels,
    float* __restrict__ partials) {
  __shared__ float rs[256];
  __shared__ float rc[256];
  const int bd = blockIdx.x;
  const int b = bd >> 10;
  const float d0 = dproj[(size_t)bd * 3 + 0] + b1[0];
  const float d1 = dproj[(size_t)bd * 3 + 1] + b1[1];
  const float d2 = dproj[(size_t)bd * 3 + 2] + b1[2];
  const float* ep = eproj + (size_t)b * H_DIM * 3;
  const int* lab = labels + (size_t)bd * 1024;

  float s = 0.f, c = 0.f;
  for (int e = threadIdx.x; e < 1024; e += 256) {
    const float l0 = d0 + ep[e * 3 + 0];
    const float l1 = d1 + ep[e * 3 + 1];
    const float l2 = d2 + ep[e * 3 + 2];
    const int lb = lab[e];
    const bool valid = (lb != -100);
    const float m = fmaxf(l0, fmaxf(l1, l2));
    const float lse =
        m + __logf(__expf(l0 - m) + __expf(l1 - m) + __expf(l2 - m));
    const int sl = valid ? lb : 0;
    const float sel = (sl == 0) ? l0 : ((sl == 1) ? l1 : l2);
    if (valid) {
      s += lse - sel;
      c += 1.0f;
    }
  }
  rs[threadIdx.x] = s;
  rc[threadIdx.x] = c;
  __syncthreads();
  for (int st = 128; st > 0; st >>= 1) {
    if (threadIdx.x < (unsigned)st) {
      rs[threadIdx.x] += rs[threadIdx.x + st];
      rc[threadIdx.x] += rc[threadIdx.x + st];
    }
    __syncthreads();
  }
  if (threadIdx.x == 0) {
    partials[(size_t)bd * 2 + 0] = rs[0];
    partials[(size_t)bd * 2 + 1] = rc[0];
  }
}

// Deterministic final reduction (fixed order every replay).
__global__ __launch_bounds__(256) void final_reduce_kernel(
    const float* __restrict__ partials, int n, float* __restrict__ out) {
  __shared__ float rs[256];
  __shared__ float rc[256];
  float s = 0.f, c = 0.f;
  for (int i = threadIdx.x; i < n; i += 256) {
    s += partials[(size_t)i * 2 + 0];
    c += partials[(size_t)i * 2 + 1];
  }
  rs[threadIdx.x] = s;
  rc[threadIdx.x] = c;
  __syncthreads();
  for (int st = 128; st > 0; st >>= 1) {
    if (threadIdx.x < (unsigned)st) {
      rs[threadIdx.x] += rs[threadIdx.x + st];
      rc[threadIdx.x] += rc[threadIdx.x + st];
    }
    __syncthreads();
  }
  if (threadIdx.x == 0) out[0] = rs[0] / fmaxf(rc[0], 1.0f);
}

extern "C" void kernel_launch(void* const* d_in, const int* in_sizes, int n_in,
                              void* d_out, int out_size, void* d_ws,
                              size_t ws_size, hipStream_t stream) {
  (void)in_sizes; (void)n_in; (void)out_size; (void)ws_size;
  const float* enc   = (const float*)d_in[0];  // [8,1024,768]
  const float* dec   = (const float*)d_in[1];  // [1,8,1024,768] == [8,1024,768]
  const float* emask = (const float*)d_in[2];  // [8,1024]
  const float* dmask = (const float*)d_in[3];  // [8,1024]
  const float* W_e   = (const float*)d_in[4];  // [1024,768]
  const float* b_e   = (const float*)d_in[5];  // [1024]
  const float* W_d   = (const float*)d_in[6];  // [1024,768]
  const float* b_d   = (const float*)d_in[7];  // [1024]
  const float* W1    = (const float*)d_in[8];  // [3,2048]
  const float* b1    = (const float*)d_in[9];  // [3]
  const int* labels  = (const int*)d_in[10];   // [8,1024,1024]
  float* out = (float*)d_out;

  float* dproj    = (float*)d_ws;              // 8*1024*3
  float* eproj    = dproj + 8 * 1024 * 3;      // 8*1024*3
  float* partials = eproj + 8 * 1024 * 3;      // 8192*2

  proj_fused_kernel<<<256, 128, 0, stream>>>(dec, W_d, b_d, W1, 0, dmask, dproj);
  proj_fused_kernel<<<256, 128, 0, stream>>>(enc, W_e, b_e, W1, 1024, emask, eproj);
  grid_nll_kernel<<<8192, 256, 0, stream>>>(dproj, eproj, b1, labels, partials);
  final_reduce_kernel<<<1, 256, 0, stream>>>(partials, 8192, out);
}